// DistillationEncoderPrototypesLoss_57801669870282
// MI455X (gfx1250) — compile-verified
//
#include <hip/hip_runtime.h>

typedef __attribute__((ext_vector_type(2))) float v2f;
typedef __attribute__((ext_vector_type(8))) float v8f;

#define BATCH   8
#define CF      256
#define HF      128
#define WF      128
#define HWF     (HF*WF)        // 16384
#define HL      512
#define WL      512
#define COLD    16
#define NPIX    (BATCH*HWF)    // 131072
#define TILES   (NPIX/16)      // 8192
#define NBLK    256
#define WPB     8              // waves per block (256 threads, wave32)
#define TPW     (TILES/(NBLK*WPB))  // 4 tiles per wave

// ---------------- kernel 0: zero the global class accumulators -------------
__global__ void k_init(float* acc) {
    if (threadIdx.x < 2*COLD) acc[threadIdx.x] = 0.0f;
}

// ---------------- kernel 1: pseudo-label ids -------------------------------
// ids[n] = 0 unless labels[b,4h,4w]==0, else argmax_ch outputs_old[b,ch,4h,4w]
__global__ void k_pseudo(const float* __restrict__ outputs_old,
                         const int*   __restrict__ labels,
                         int*         __restrict__ ids) {
    int n = blockIdx.x * blockDim.x + threadIdx.x;
    if (n >= NPIX) return;
    int b  = n >> 14;
    int hw = n & (HWF-1);
    int h4 = (hw >> 7) << 2;
    int w4 = (hw & 127) << 2;
    long pix = (long)h4 * WL + w4;
    int lbl = labels[(long)b * HL * WL + pix];
    int id = 0;
    if (lbl == 0) {
        long base = (long)b * COLD * HL * WL + pix;
        float best = outputs_old[base];
        int bi = 0;
        #pragma unroll
        for (int ch = 1; ch < COLD; ++ch) {
            float v = outputs_old[base + (long)ch * HL * WL];
            if (v > best) { best = v; bi = ch; }   // first-max tie rule, like argmax
        }
        id = bi;
    }
    ids[n] = id;
}

// ---------------- kernel 2: WMMA cross-term + per-class accumulation -------
__global__ __launch_bounds__(256) void
k_main(const float* __restrict__ features,
       const float* __restrict__ prototypes,
       const int*   __restrict__ ids,
       float*       __restrict__ gsums,
       float*       __restrict__ gcounts) {
    __shared__ float protoT[CF][COLD];       // 16 KB, B-operand source
    __shared__ float ssqp[COLD];             // ||proto_k||^2
    __shared__ float csums[COLD];
    __shared__ float ccnts[COLD];
    __shared__ float gbuf[WPB][16][17];      // padded G spill, 8.5 KB

    const int tid  = threadIdx.x;
    const int lane = tid & 31;
    const int wave = tid >> 5;
    const int lo   = lane & 15;
    const int hi   = lane >> 4;              // half-wave: 0 -> K 0/1, 1 -> K 2/3

    // stage transposed prototypes into LDS
    for (int i = tid; i < CF * COLD; i += 256) {
        int c = i >> 4, k = i & 15;
        protoT[c][k] = prototypes[k * CF + c];
    }
    if (tid < COLD) {
        float s = 0.0f;
        for (int c = 0; c < CF; ++c) {
            float p = prototypes[tid * CF + c];
            s += p * p;
        }
        ssqp[tid]  = s;
        csums[tid] = 0.0f;
        ccnts[tid] = 0.0f;
    }
    __syncthreads();

    const int gw = blockIdx.x * WPB + wave;

    for (int i = 0; i < TPW; ++i) {
        int t   = gw * TPW + i;
        int n0  = t << 4;                    // 16 pixels, same image
        int b   = n0 >> 14;
        int hw0 = n0 & (HWF-1);
        const float* fbase = features + (size_t)b * CF * HWF + hw0 + lo;

        v8f acc = {0.f,0.f,0.f,0.f,0.f,0.f,0.f,0.f};
        float ssq = 0.0f;

        #pragma unroll 4
        for (int c0 = 0; c0 < CF; c0 += 4) {
            int cA = c0 + (hi << 1);
            float a0 = fbase[(size_t)cA       * HWF];
            float a1 = fbase[(size_t)(cA + 1) * HWF];
            ssq += a0 * a0 + a1 * a1;
            v2f a  = {a0, a1};
            v2f bb = {protoT[cA][lo], protoT[cA + 1][lo]};
            // D = A(16x4 f32) x B(4x16 f32) + C, full f32 precision
            acc = __builtin_amdgcn_wmma_f32_16x16x4_f32(
                false, a, false, bb, (short)0, acc, false, false);
        }

        // spill G tile (C/D layout: VGPR r -> M=r (+8 for hi half), N=lo)
        int rbase = hi << 3;
        #pragma unroll
        for (int r = 0; r < 8; ++r)
            gbuf[wave][rbase + r][lo] = acc[r];
        asm volatile("s_wait_dscnt 0x0" ::: "memory");

        float ssq_tot = ssq + __shfl_xor(ssq, 16, 32);  // combine K halves

        if (lane < 16) {
            int   idm = ids[n0 + lane];
            float g   = gbuf[wave][lane][idm];
            float pp  = (ssq_tot - 2.0f * g + ssqp[idm]) * (1.0f / CF);
            atomicAdd(&csums[idm], pp);
            atomicAdd(&ccnts[idm], 1.0f);
        }
    }

    __syncthreads();
    if (tid < COLD) {
        atomicAdd(&gsums[tid],   csums[tid]);
        atomicAdd(&gcounts[tid], ccnts[tid]);
    }
}

// ---------------- kernel 3: finalize --------------------------------------
__global__ void k_final(const float* __restrict__ gsums,
                        const float* __restrict__ gcounts,
                        float* __restrict__ out) {
    int lane = threadIdx.x;
    float v = 0.0f;
    if (lane >= 1 && lane < COLD) {          // class 0 excluded
        float c = gcounts[lane];
        if (c > 0.0f) v = gsums[lane] / c;
    }
    #pragma unroll
    for (int off = 16; off > 0; off >>= 1)
        v += __shfl_down(v, off, 32);
    if (lane == 0) out[0] = v;
}

// ---------------- launcher -------------------------------------------------
extern "C" void kernel_launch(void* const* d_in, const int* in_sizes, int n_in,
                              void* d_out, int out_size, void* d_ws, size_t ws_size,
                              hipStream_t stream) {
    // inputs: 0 outputs, 1 outputs_old, 2 features, 3 features_old,
    //         4 labels, 5 prototypes, 6 classes_old, 7 incremental_step
    const float* outputs_old = (const float*)d_in[1];
    const float* features    = (const float*)d_in[2];
    const int*   labels      = (const int*)d_in[4];
    const float* prototypes  = (const float*)d_in[5];
    float* out = (float*)d_out;

    float* gsums   = (float*)d_ws;           // [16]
    float* gcounts = gsums + COLD;           // [16]
    int*   ids     = (int*)(gcounts + COLD); // [131072]

    k_init  <<<1, 64, 0, stream>>>(gsums);
    k_pseudo<<<NPIX / 256, 256, 0, stream>>>(outputs_old, labels, ids);
    k_main  <<<NBLK, 256, 0, stream>>>(features, prototypes, ids, gsums, gcounts);
    k_final <<<1, 32, 0, stream>>>(gsums, gcounts, out);
}